// CrossEntropyLossWithGaussianSmoothedLabels_35390530519630
// MI455X (gfx1250) — compile-verified
//
#include <hip/hip_runtime.h>
#include <math.h>

#define NC 722
#define NC2 361               // NC/2 float2 per row (rows always 8B-aligned)
#define ROWS_PER_BLOCK 8      // 8 waves of 32 per block
#define K2 12                 // ceil(361/32)

typedef __attribute__((ext_vector_type(2))) float v2f;
typedef __attribute__((ext_vector_type(8))) float v8f;

#define LOG2E 1.44269504088896340736f
#define LN2   0.69314718055994530942f

// Exact 32-lane f32 sum, broadcast to all lanes, via two V_WMMA_F32_16X16X4_F32.
// B = all-ones is layout-independent; A = (x, 0) => D[m][j] = x[m] + x[m+16].
// Per-lane sum of the 8 D VGPRs folds rows 0-7 (lanes 0-15) / 8-15 (lanes 16-31);
// a second ones-WMMA adds the halves and replicates the total in every element.
__device__ __forceinline__ float wave_sum32(float x) {
  v2f a;  a[0] = x;    a[1] = 0.0f;
  v2f b;  b[0] = 1.0f; b[1] = 1.0f;
  v8f cz = {0.f, 0.f, 0.f, 0.f, 0.f, 0.f, 0.f, 0.f};
  v8f d = __builtin_amdgcn_wmma_f32_16x16x4_f32(false, a, false, b, (short)0, cz,
                                                false, false);
  float g = ((d[0] + d[1]) + (d[2] + d[3])) + ((d[4] + d[5]) + (d[6] + d[7]));
  v2f a2; a2[0] = g;   a2[1] = 0.0f;
  v8f d2 = __builtin_amdgcn_wmma_f32_16x16x4_f32(false, a2, false, b, (short)0, cz,
                                                 false, false);
  return d2[0];
}

__global__ __launch_bounds__(256)
void ce_gauss_rows(const float* __restrict__ pred,
                   const int* __restrict__ target,
                   float* __restrict__ partial) {
  const int lane = threadIdx.x & 31;
  const int wid  = threadIdx.x >> 5;
  const int row  = blockIdx.x * ROWS_PER_BLOCK + wid;
  const float*  base  = pred + (size_t)row * NC;
  const float2* base2 = reinterpret_cast<const float2*>(base);  // 8B aligned

  // Stream the whole row into registers with coalesced b64 loads (one HBM pass).
  float2 v[K2];
#pragma unroll
  for (int k = 0; k < K2 - 1; ++k) v[k] = base2[lane + 32 * k];
  {
    const int idx = lane + 32 * (K2 - 1);
    const float2 t = base2[idx < NC2 ? idx : NC2 - 1];  // clamped, branchless
    v[K2 - 1].x = (idx < NC2) ? t.x : -INFINITY;        // -inf: identity for max/exp
    v[K2 - 1].y = (idx < NC2) ? t.y : -INFINITY;
  }

  // Row max (registers + 5 xor-shuffles).
  float m = fmaxf(v[0].x, v[0].y);
#pragma unroll
  for (int k = 1; k < K2; ++k) m = fmaxf(m, fmaxf(v[k].x, v[k].y));
#pragma unroll
  for (int off = 16; off > 0; off >>= 1) m = fmaxf(m, __shfl_xor(m, off, 32));

  // sum exp(v - m) = sum exp2(v*log2e - m*log2e): 1 FMA + 1 v_exp_f32 + 1 add.
  const float negml = -m * LOG2E;
  float s = 0.0f;
#pragma unroll
  for (int k = 0; k < K2; ++k) {
    s += __builtin_amdgcn_exp2f(fmaf(v[k].x, LOG2E, negml));
    s += __builtin_amdgcn_exp2f(fmaf(v[k].y, LOG2E, negml));
  }
  const float S   = wave_sum32(s);
  const float lse = fmaf(LN2, __builtin_amdgcn_logf(S), m);

  // Smoothed-label part. Lanes 0..6 own candidate c = t + (lane-3).
  // The final (overwrite-resolved) weight at any in-range c is exactly
  // DECAYS[|c-t|]: the unclipped write at d=|c-t| always lands on c, and any
  // smaller d can reach c only by clipping, which would require d > |c-t|.
  // So the weight is a pure per-lane constant, masked by range validity.
  const int t = target[row];
  const int c = t + (lane - 3);
  const bool valid = (lane < 7) && (c >= 0) && (c < NC);
  const int  ad = (lane < 3) ? (3 - lane) : (lane - 3);   // |c - t| for lanes 0..6
  const float w = (ad == 0) ? 0.77880078307140486825f     // exp(-1/4)
                : (ad == 1) ? 0.60653065971263342360f     // exp(-1/2)
                : (ad == 2) ? 0.36787944117144232160f     // exp(-1)
                            : 0.13533528323661269189f;    // exp(-2)
  const float p  = base[valid ? c : 0];                   // row is hot in cache
  const float wv = valid ? w : 0.0f;

  // loss = SW*lse - DOT = sum_lanes w*(lse - p): single fused WMMA reduction.
  const float loss = wave_sum32(wv * (lse - p));

  __shared__ float waveloss[ROWS_PER_BLOCK];
  if (lane == 0) waveloss[wid] = loss;
  __syncthreads();
  if (threadIdx.x == 0) {
    float acc = 0.0f;
#pragma unroll
    for (int i = 0; i < ROWS_PER_BLOCK; ++i) acc += waveloss[i];  // fixed order
    partial[blockIdx.x] = acc;
  }
}

__global__ __launch_bounds__(256)
void ce_gauss_reduce(const float* __restrict__ partial, float* __restrict__ out,
                     int n, float inv_count) {
  __shared__ float sm[256];
  float s = 0.0f;
  for (int i = threadIdx.x; i < n; i += 256) s += partial[i];  // fixed order
  sm[threadIdx.x] = s;
  __syncthreads();
#pragma unroll
  for (int st = 128; st > 0; st >>= 1) {                       // fixed tree
    if (threadIdx.x < st) sm[threadIdx.x] += sm[threadIdx.x + st];
    __syncthreads();
  }
  if (threadIdx.x == 0) out[0] = sm[0] * inv_count;
}

extern "C" void kernel_launch(void* const* d_in, const int* in_sizes, int n_in,
                              void* d_out, int out_size, void* d_ws, size_t ws_size,
                              hipStream_t stream) {
  const float* pred   = (const float*)d_in[0];   // [16, 4096, 722] f32
  const int*   target = (const int*)d_in[1];     // [16, 4096] i32
  float* out = (float*)d_out;                    // scalar f32
  float* wsf = (float*)d_ws;                     // 8192 partials (32 KB)

  const int rows   = in_sizes[1];                // 65536
  const int blocks = rows / ROWS_PER_BLOCK;      // 8192

  ce_gauss_rows<<<blocks, 256, 0, stream>>>(pred, target, wsf);
  ce_gauss_reduce<<<1, 256, 0, stream>>>(wsf, out, blocks, 1.0f / (float)rows);
}